// HSMMModel_69767448756747
// MI455X (gfx1250) — compile-verified
//
#include <hip/hip_runtime.h>

#define L_DIM 11
#define B_DIM 16
#define T_DIM 512
#define Z_DIM 256
#define V_DIM 50000
#define NEG_INF_F (-1000000.0f)
#define NCHUNK 125
#define CHUNK 400

typedef float v2f __attribute__((ext_vector_type(2)));
typedef float v8f __attribute__((ext_vector_type(8)));

// workspace layout (float offsets)
#define WS_P     0                              // Z*Z   = 65536  (P = exp(log_softmax(z_per_z)))
#define WS_LSE   (WS_P + Z_DIM * Z_DIM)         // L*Z   = 2816   (logsumexp over V)
#define WS_LT    (WS_LSE + L_DIM * Z_DIM)       // L*Z   = 2816   (l_term[l][z])
#define WS_MSG0  (WS_LT + L_DIM * Z_DIM)        // Z             (log_z[0][z'])
#define WS_PART  (WS_MSG0 + Z_DIM)              // L*NCHUNK*Z partial lse

// ---------------------------------------------------------------------------
// Kernel 1: partial logsumexp over V chunks.  grid (NCHUNK, L), block 256.
// emb layout (L, V, Z), z contiguous -> coalesced loads per v-row.
__global__ __launch_bounds__(256) void k_lse_partial(const float* __restrict__ emb,
                                                     float* __restrict__ part) {
    const int c = blockIdx.x;
    const int l = blockIdx.y;
    const int z = threadIdx.x;
    const float* base = emb + (long)l * V_DIM * Z_DIM + z;
    const int v0 = c * CHUNK;
    float m = -3.0e38f, s = 0.0f;
    for (int v = v0; v < v0 + CHUNK; ++v) {
        float x = base[(long)v * Z_DIM];
        float nm = fmaxf(m, x);
        s = s * __expf(m - nm) + __expf(x - nm);
        m = nm;
    }
    part[((l * NCHUNK + c) * Z_DIM) + z] = m + __logf(s);
}

// Kernel 2: reduce NCHUNK partial lse's -> lse[l][z].  grid L, block 256.
__global__ __launch_bounds__(256) void k_lse_reduce(const float* __restrict__ part,
                                                    float* __restrict__ lse) {
    const int l = blockIdx.x;
    const int z = threadIdx.x;
    float m = -3.0e38f, s = 0.0f;
    for (int c = 0; c < NCHUNK; ++c) {
        float p = part[((l * NCHUNK + c) * Z_DIM) + z];
        float nm = fmaxf(m, p);
        s = s * __expf(m - nm) + __expf(p - nm);
        m = nm;
    }
    lse[l * Z_DIM + z] = m + __logf(s);
}

// Kernel 3: row-wise softmax of z_per_z -> P (probabilities); also msg0 = log_z[0][:].
// grid Z, block 256.
__global__ __launch_bounds__(256) void k_row_softmax(const float* __restrict__ zz,
                                                     float* __restrict__ P,
                                                     float* __restrict__ msg0) {
    __shared__ float red[Z_DIM];
    const int r = blockIdx.x;
    const int j = threadIdx.x;
    float x = zz[r * Z_DIM + j];
    red[j] = x;
    __syncthreads();
    for (int s = 128; s > 0; s >>= 1) {
        if (j < s) red[j] = fmaxf(red[j], red[j + s]);
        __syncthreads();
    }
    float m = red[0];
    __syncthreads();
    red[j] = __expf(x - m);
    __syncthreads();
    for (int s = 128; s > 0; s >>= 1) {
        if (j < s) red[j] += red[j + s];
        __syncthreads();
    }
    float ssum = red[0];
    P[r * Z_DIM + j] = __expf(x - m) / ssum;
    if (r == 0) msg0[j] = (x - m) - __logf(ssum);
}

// Kernel 4: l_term[l][z] = l_per_z[z][l] - logsumexp_l' l_per_z[z][l'].
// (the MIN_L NEG_INF pad contributes exp(-1e6 - m) == 0 in f32, identical to ref)
__global__ __launch_bounds__(256) void k_lterm(const float* __restrict__ lp,
                                               float* __restrict__ lt) {
    const int z = threadIdx.x;
    float m = -3.0e38f;
    for (int l = 0; l < L_DIM; ++l) m = fmaxf(m, lp[z * L_DIM + l]);
    float s = 0.0f;
    for (int l = 0; l < L_DIM; ++l) s += __expf(lp[z * L_DIM + l] - m);
    float ls = m + __logf(s);
    for (int l = 0; l < L_DIM; ++l) lt[l * Z_DIM + z] = lp[z * L_DIM + l] - ls;
}

// ---------------------------------------------------------------------------
// Kernel 5: the sequential HSMM scan.  ONE workgroup, 512 threads = 16 waves.
// Wave w owns the 16-wide z' tile [16w, 16w+16) of the (16x256)@(256x256) GEMM,
// done with V_WMMA_F32_16X16X4_F32, K=256 -> 64 WMMAs/wave/step.
__global__ __launch_bounds__(512) void k_dp(const int* __restrict__ ng,
                                            const float* __restrict__ emb,
                                            const int* __restrict__ xlen,
                                            const float* __restrict__ ws,
                                            float* __restrict__ out) {
    __shared__ float msgbuf[L_DIM][B_DIM][Z_DIM];   // circular window of msgs (176 KB)
    __shared__ float abuf[B_DIM][Z_DIM];            // beta, then exp(beta - mb)
    __shared__ float lt_sh[L_DIM][Z_DIM];           // l_term
    __shared__ float ltl_sh[L_DIM][Z_DIM];          // l_term - lse (folded)
    __shared__ float mb_sh[B_DIM];

    const int t = threadIdx.x;
    const int lane = t & 31;
    const int w = t >> 5;                 // wave id == batch row it reduces
    const float* Pm   = ws + WS_P;
    const float* lseg = ws + WS_LSE;
    const float* ltg  = ws + WS_LT;
    const float* msg0 = ws + WS_MSG0;

    // stage l_term / (l_term - lse)
    for (int i = t; i < L_DIM * Z_DIM; i += 512) {
        float ltv = ltg[i];
        lt_sh[i / Z_DIM][i % Z_DIM] = ltv;
        ltl_sh[i / Z_DIM][i % Z_DIM] = ltv - lseg[i];
    }
    // window filled with NEG_INF (reference uses finite -1e6 fill)
    for (int i = t; i < L_DIM * B_DIM * Z_DIM; i += 512)
        ((float*)msgbuf)[i] = NEG_INF_F;
    __syncthreads();
    // slot 0 holds msg_0 = log_z[INIT_STATE][:] broadcast over batch
    for (int i = t; i < B_DIM * Z_DIM; i += 512)
        msgbuf[0][i >> 8][i & 255] = msg0[i & 255];

    // loop-invariant B fragments: P tile for this wave's 16 columns.
    // B (4x16 KxN) layout guess mirrors documented A 16x4: VGPR0 = K {0|2}, VGPR1 = K {1|3}.
    v2f bf[64];
    {
        const int hi = lane >> 4, j = lane & 15;
        const int col = (w << 4) + j;
#pragma unroll
        for (int kk = 0; kk < 64; ++kk) {
            int k0 = 4 * kk + 2 * hi;
            bf[kk].x = Pm[k0 * Z_DIM + col];
            bf[kk].y = Pm[(k0 + 1) * Z_DIM + col];
        }
    }
    const int xl = xlen[w];               // wave-uniform
    if (lane == 0 && xl == 0) out[w] = 0.0f;   // betas[0] = init -> logsumexp = 0
    __syncthreads();

    const int b = w;                      // phase-1 batch row (wave-uniform)
    const int zt = lane << 3;             // 8 contiguous z per thread

    for (int n = 1; n < T_DIM; ++n) {
        // ---- phase 1: beta[b][z] = logsumexp_l (win + l_term + e) -------------
        int idr[L_DIM];
#pragma unroll
        for (int l = 0; l < L_DIM; ++l) {
            int tsrc = n - 1 - l;
            idr[l] = (tsrc >= 0) ? ng[(l * B_DIM + b) * T_DIM + tsrc] : 1; // pad==id 1 path
        }
        float mx[8], sacc[8];
#pragma unroll
        for (int i = 0; i < 8; ++i) { mx[i] = -3.0e38f; sacc[i] = 0.0f; }

        for (int pass = 0; pass < 2; ++pass) {
#pragma unroll
            for (int l = 0; l < L_DIM; ++l) {
                const int sl = (n + 10 - l) % 11;   // slot of msg_{n-1-l}
                const float* wrow = &msgbuf[sl][b][zt];
                const float* ltp  = &lt_sh[l][zt];
                const int id = idr[l];
                if (id > 1) {
                    const float* ltlp = &ltl_sh[l][zt];
                    const float* ep = emb + ((long)(l * V_DIM + id)) * Z_DIM + zt;
#pragma unroll
                    for (int i = 0; i < 8; ++i) {
                        float tv = wrow[i] + ltlp[i] + ep[i];
                        if (pass == 0) mx[i] = fmaxf(mx[i], tv);
                        else sacc[i] += __expf(tv - mx[i]);
                    }
                } else if (id == 0) {
#pragma unroll
                    for (int i = 0; i < 8; ++i) {
                        float tv = wrow[i] + ltp[i];
                        if (pass == 0) mx[i] = fmaxf(mx[i], tv);
                        else sacc[i] += __expf(tv - mx[i]);
                    }
                } else {
#pragma unroll
                    for (int i = 0; i < 8; ++i) {
                        float tv = wrow[i] + ltp[i] + NEG_INF_F;
                        if (pass == 0) mx[i] = fmaxf(mx[i], tv);
                        else sacc[i] += __expf(tv - mx[i]);
                    }
                }
            }
        }
#pragma unroll
        for (int i = 0; i < 8; ++i) abuf[b][zt + i] = mx[i] + __logf(sacc[i]);
        __syncthreads();

        // ---- phase 2: row max/sum, a = exp(beta - mb), capture output ---------
        float lm = -3.0e38f;
#pragma unroll
        for (int i = 0; i < 8; ++i) lm = fmaxf(lm, abuf[w][lane + 32 * i]);
        for (int off = 16; off > 0; off >>= 1) lm = fmaxf(lm, __shfl_xor(lm, off, 32));
        float lsum = 0.0f;
#pragma unroll
        for (int i = 0; i < 8; ++i) {
            float av = __expf(abuf[w][lane + 32 * i] - lm);
            abuf[w][lane + 32 * i] = av;
            lsum += av;
        }
        for (int off = 16; off > 0; off >>= 1) lsum += __shfl_xor(lsum, off, 32);
        if (lane == 0) {
            mb_sh[w] = lm;
            if (xl == n) out[w] = lm + __logf(lsum);  // logsumexp_z beta_n[b]
        }
        __syncthreads();

        // ---- phase 3: D = A @ P_tile via f32 WMMA, K = 256 --------------------
        v8f acc = {0.f, 0.f, 0.f, 0.f, 0.f, 0.f, 0.f, 0.f};
        const int am = lane & 15;
        const int ac = (lane >> 4) << 1;    // A 16x4: VGPR0 = K {0|2}, VGPR1 = K {1|3}
#pragma unroll
        for (int kk = 0; kk < 64; ++kk) {
            v2f A;
            A.x = abuf[am][4 * kk + ac];
            A.y = abuf[am][4 * kk + ac + 1];
            acc = __builtin_amdgcn_wmma_f32_16x16x4_f32(false, A, false, bf[kk],
                                                        (short)0, acc, false, false);
        }

        // ---- phase 4: msg_n = mb + log(D) into circular slot n%11 -------------
        const int slotn = n % 11;
        const int col = (w << 4) | am;
        const int hb = (lane >> 4) << 3;    // D: VGPR r -> M = r (+8 for hi half-wave)
#pragma unroll
        for (int r = 0; r < 8; ++r) {
            int mr = r + hb;
            msgbuf[slotn][mr][col] = mb_sh[mr] + __logf(acc[r]);
        }
        __syncthreads();
    }
}

// ---------------------------------------------------------------------------
extern "C" void kernel_launch(void* const* d_in, const int* in_sizes, int n_in,
                              void* d_out, int out_size, void* d_ws, size_t ws_size,
                              hipStream_t stream) {
    // inputs: x(unused), x_lengths, ngram_ids, emb, z_per_z, l_per_z
    const int*   xlen = (const int*)d_in[1];
    const int*   ng   = (const int*)d_in[2];
    const float* emb  = (const float*)d_in[3];
    const float* zz   = (const float*)d_in[4];
    const float* lp   = (const float*)d_in[5];
    float* out = (float*)d_out;
    float* wsf = (float*)d_ws;

    k_lse_partial<<<dim3(NCHUNK, L_DIM), 256, 0, stream>>>(emb, wsf + WS_PART);
    k_lse_reduce <<<L_DIM, 256, 0, stream>>>(wsf + WS_PART, wsf + WS_LSE);
    k_row_softmax<<<Z_DIM, 256, 0, stream>>>(zz, wsf + WS_P, wsf + WS_MSG0);
    k_lterm      <<<1, 256, 0, stream>>>(lp, wsf + WS_LT);
    k_dp         <<<1, 512, 0, stream>>>(ng, emb, xlen, wsf, out);
}